// PhaseLoss_46866683134089
// MI455X (gfx1250) — compile-verified
//
#include <hip/hip_runtime.h>
#include <math.h>

typedef float v2f __attribute__((ext_vector_type(2)));
typedef float v8f __attribute__((ext_vector_type(8)));

#define NFFT     512
#define HOP      128
#define NFREQ    257          // n_fft/2 + 1
#define FPAD     272          // 17 freq tiles of 16
#define NPAIRS   256          // NFFT/2 K-pairs in twiddle tables
#define NFRAMES  32769        // T/HOP + 1
#define NSAMP    4194304      // 2^22
#define FTILES   2049         // ceil(32769/16)
#define SPAN     2432         // 15*HOP + NFFT : samples covered by 16 frames
#define TWO_PI_F     6.2831853071795864769f
#define INV_TWO_PI_F 0.15915494309189533577f

// ---------------- workspace layout ----------------
// [0, 24)               : double acc[3]
// [64, +256*272*8)      : Wc2  float2 {cos(2p,f), cos(2p+1,f)}
// next 256*272*8        : Ws2  float2 {-sin(2p,f), -sin(2p+1,f)}
// next 257*32769*4      : PR (phase of yd, [f][t])
// next 257*32769*4      : PG (phase of td)
#define W_OFF    64
#define WTAB_SZ  ((size_t)NPAIRS * FPAD * 8)
#define PH_ELEMS ((long)NFREQ * NFRAMES)

__device__ __forceinline__ float anti_wrap(float x) {
    // |x - round(x/2pi)*2pi| ; rintf = round-half-even, matches jnp.round
    return fabsf(fmaf(-rintf(x * INV_TWO_PI_F), TWO_PI_F, x));
}

// ------ kernel 0: de-interleaved twiddle tables + accumulator init -------
__global__ void pl_setup_kernel(v2f* __restrict__ Wc2, v2f* __restrict__ Ws2,
                                double* __restrict__ acc) {
    int i = blockIdx.x * blockDim.x + threadIdx.x;
    if (i == 0) { acc[0] = 0.0; acc[1] = 0.0; acc[2] = 0.0; }
    if (i < NPAIRS * FPAD) {
        int pair = i / FPAD;
        int f    = i % FPAD;
        int k0   = pair * 2;
        // exp(-2pi i f k / 512): exact argument reduction via (f*k mod 512)
        int r0 = (k0 * f) & (NFFT - 1);
        int r1 = ((k0 + 1) * f) & (NFFT - 1);
        float s0, c0, s1, c1;
        sincosf((float)r0 * (TWO_PI_F / (float)NFFT), &s0, &c0);
        sincosf((float)r1 * (TWO_PI_F / (float)NFFT), &s1, &c1);
        v2f qc, qs;
        qc.x = c0;  qc.y = c1;
        qs.x = -s0; qs.y = -s1;   // Im weight pre-negated: Im = sum x*(-sin)
        Wc2[i] = qc;
        Ws2[i] = qs;
    }
}

// ---------------- kernel 1: DFT-as-GEMM phase via WMMA f32 ----------------
// block = (32, 17): 17 waves, wave w handles freq tile [16w, 16w+16)
// grid  = (2049, 2): frame tiles x {yd, td}
__global__ __launch_bounds__(544)
void pl_dft_phase_kernel(const float* __restrict__ yd, const float* __restrict__ td,
                         const v2f* __restrict__ Wc2, const v2f* __restrict__ Ws2,
                         float* __restrict__ PR, float* __restrict__ PG) {
    __shared__ float sx[SPAN];

    const int lane = threadIdx.x;          // 0..31
    const int wave = threadIdx.y;          // 0..16
    const int t0   = blockIdx.x * 16;      // first frame of tile
    const int sig  = blockIdx.y;

    const float* __restrict__ x = sig ? td : yd;
    float* __restrict__ P       = sig ? PG : PR;

    // cooperative staging of the overlapping 16-frame span, reflect-padded
    const long j0 = (long)t0 * HOP - (NFFT / 2);
    for (int i = wave * 32 + lane; i < SPAN; i += 17 * 32) {
        long j = j0 + i;
        if (j < 0) j = -j;                         // left reflect (no edge repeat)
        if (j >= NSAMP) j = 2L * NSAMP - 2 - j;    // right reflect
        sx[i] = x[j];
    }
    __syncthreads();

    const int f0    = wave * 16;
    const int m     = lane & 15;            // A-matrix row (frame within tile)
    const int khalf = (lane >> 4) << 1;     // 0 (lanes 0-15) or 2 (lanes 16-31)
    const int kh2   = lane >> 4;            // khalf/2: pair-index offset
    const int n     = f0 + (lane & 15);     // B-matrix column (frequency)

    v8f cre = {0.f, 0.f, 0.f, 0.f, 0.f, 0.f, 0.f, 0.f};
    v8f cim = {0.f, 0.f, 0.f, 0.f, 0.f, 0.f, 0.f, 0.f};

    // induction pointers -> immediate-offset loads after unrolling
    const v2f* pc = Wc2 + kh2 * FPAD + n;          // cos pair for this lane
    const v2f* ps = Ws2 + kh2 * FPAD + n;          // -sin pair for this lane
    const v2f* pa = (const v2f*)(sx + m * HOP + khalf);  // 8B-aligned LDS pair

    #pragma unroll 8
    for (int k = 0; k < NFFT; k += 4) {
        // A 16x4 f32 fragment: {x[kk], x[kk+1]} straight from LDS
        const v2f a  = *pa;
        // B 4x16 f32 fragments: aligned b64 loads, no shuffles
        const v2f bc = *pc;
        const v2f bs = *ps;
        pa += 2;           // +4 floats
        pc += 2 * FPAD;    // next k-pair row
        ps += 2 * FPAD;
        // D = A*B + C, full fp32
        cre = __builtin_amdgcn_wmma_f32_16x16x4_f32(false, a, false, bc,
                                                    (short)0, cre, false, false);
        cim = __builtin_amdgcn_wmma_f32_16x16x4_f32(false, a, false, bs,
                                                    (short)0, cim, false, false);
    }

    // C/D 16x16 f32 layout: VGPR v -> M = v + 8*(lane>=16), N = lane&15
    const int f     = f0 + (lane & 15);
    const int tbase = t0 + ((lane >> 4) << 3);
    if (f < NFREQ) {
        #pragma unroll
        for (int v = 0; v < 8; ++v) {
            const int t = tbase + v;
            if (t < NFRAMES) {
                // spec = Re + i*Im (Im accumulated with -sin) -> atan2(Im, Re)
                P[(long)f * NFRAMES + t] = atan2f(cim[v], cre[v]);
            }
        }
    }
}

// ---------------- kernel 2: fused IP/GD/IAF loss reduction ----------------
// grid = (ceil(NFRAMES/256), NFREQ), block = 256
__global__ void pl_loss_kernel(const float* __restrict__ PR, const float* __restrict__ PG,
                               double* __restrict__ acc) {
    const int tx = threadIdx.x;
    const int t  = blockIdx.x * 256 + tx;
    const int f  = blockIdx.y;

    float s_ip = 0.f, s_gd = 0.f, s_iaf = 0.f;
    if (t < NFRAMES) {
        const long idx = (long)f * NFRAMES + t;
        const float pr  = PR[idx];
        const float pg  = PG[idx];
        const float prf = f ? PR[idx - NFRAMES] : 0.f;   // p[f-1][t], p[-1]=0
        const float pgf = f ? PG[idx - NFRAMES] : 0.f;
        const float prt = t ? PR[idx - 1]       : 0.f;   // p[f][t-1], p[·][-1]=0
        const float pgt = t ? PG[idx - 1]       : 0.f;
        s_ip  = anti_wrap(pr - pg);
        s_gd  = anti_wrap((prf - pr) - (pgf - pg));      // gd: out[f]=p[f-1]-p[f]
        s_iaf = anti_wrap((prt - pr) - (pgt - pg));      // iaf along time
    }

    __shared__ float r0[256], r1[256], r2[256];
    r0[tx] = s_ip; r1[tx] = s_gd; r2[tx] = s_iaf;
    __syncthreads();
    for (int s = 128; s > 0; s >>= 1) {
        if (tx < s) { r0[tx] += r0[tx + s]; r1[tx] += r1[tx + s]; r2[tx] += r2[tx + s]; }
        __syncthreads();
    }
    if (tx == 0) {
        atomicAdd(&acc[0], (double)r0[0]);
        atomicAdd(&acc[1], (double)r1[0]);
        atomicAdd(&acc[2], (double)r2[0]);
    }
}

// ---------------- kernel 3: finalize means ----------------
__global__ void pl_finalize_kernel(const double* __restrict__ acc, float* __restrict__ out) {
    const int i = threadIdx.x;
    if (i < 3) out[i] = (float)(acc[i] / (double)PH_ELEMS);
}

extern "C" void kernel_launch(void* const* d_in, const int* in_sizes, int n_in,
                              void* d_out, int out_size, void* d_ws, size_t ws_size,
                              hipStream_t stream) {
    const float* yd = (const float*)d_in[0];
    const float* td = (const float*)d_in[1];
    float* out = (float*)d_out;

    char* ws    = (char*)d_ws;
    double* acc = (double*)ws;
    v2f* Wc2    = (v2f*)(ws + W_OFF);
    v2f* Ws2    = (v2f*)(ws + W_OFF + WTAB_SZ);
    float* PR   = (float*)(ws + W_OFF + 2 * WTAB_SZ);
    float* PG   = PR + PH_ELEMS;

    // 0) twiddle tables + accumulator init
    pl_setup_kernel<<<(NPAIRS * FPAD + 255) / 256, 256, 0, stream>>>(Wc2, Ws2, acc);

    // 1) WMMA f32 DFT -> phases for both signals
    dim3 blk(32, 17);
    dim3 grd(FTILES, 2);
    pl_dft_phase_kernel<<<grd, blk, 0, stream>>>(yd, td, Wc2, Ws2, PR, PG);

    // 2) fused three-loss reduction
    dim3 lgrd((NFRAMES + 255) / 256, NFREQ);
    pl_loss_kernel<<<lgrd, 256, 0, stream>>>(PR, PG, acc);

    // 3) means
    pl_finalize_kernel<<<1, 32, 0, stream>>>(acc, out);
}